// EST_Net_77171972374909
// MI455X (gfx1250) — compile-verified
//
#include <hip/hip_runtime.h>
#include <hip/hip_bf16.h>

typedef __attribute__((ext_vector_type(2))) float v2f;
typedef __attribute__((ext_vector_type(8))) float v8f;

#define CB   9
#define HH   260
#define WW   346
#define HID  30
#define WH   (WW*HH)          /* 89960  */
#define WHC  (WH*CB)          /* 809640 */

// ---------------------------------------------------------------------------
// Kernel 1: per-batch segment max of t (batch in events[:,4], t in events[:,2])
// Local max in 4 registers -> wave butterfly -> uint-bit atomicMax (t >= 0).
// ---------------------------------------------------------------------------
__global__ __launch_bounds__(256) void est_tmax_kernel(
    const float* __restrict__ events, int n, unsigned* __restrict__ tmax_bits)
{
    int tid    = blockIdx.x * blockDim.x + threadIdx.x;
    int stride = gridDim.x * blockDim.x;
    float m0 = 0.f, m1 = 0.f, m2 = 0.f, m3 = 0.f;
    for (int i = tid; i < n; i += stride) {
        float t = events[(long)i * 5 + 2];
        int   b = (int)events[(long)i * 5 + 4];
        m0 = (b == 0) ? fmaxf(m0, t) : m0;
        m1 = (b == 1) ? fmaxf(m1, t) : m1;
        m2 = (b == 2) ? fmaxf(m2, t) : m2;
        m3 = (b == 3) ? fmaxf(m3, t) : m3;
    }
    #pragma unroll
    for (int mask = 1; mask < 32; mask <<= 1) {
        m0 = fmaxf(m0, __shfl_xor(m0, mask));
        m1 = fmaxf(m1, __shfl_xor(m1, mask));
        m2 = fmaxf(m2, __shfl_xor(m2, mask));
        m3 = fmaxf(m3, __shfl_xor(m3, mask));
    }
    if ((threadIdx.x & 31) == 0) {
        atomicMax(tmax_bits + 0, __float_as_uint(m0));
        atomicMax(tmax_bits + 1, __float_as_uint(m1));
        atomicMax(tmax_bits + 2, __float_as_uint(m2));
        atomicMax(tmax_bits + 3, __float_as_uint(m3));
    }
}

// ---------------------------------------------------------------------------
// Kernel 2: main. 16 events per wave (duplicated in lanes l and l+16).
// Layer 2 via V_WMMA_F32_16X16X4_F32, W2 held entirely in register fragments.
// Layer-3 reduction: tree-fold butterfly (15 shfl / bin, result lands in the
// scattering lane automatically).
// ---------------------------------------------------------------------------
__global__ __launch_bounds__(256) void est_main_kernel(
    const float* __restrict__ events,
    const float* __restrict__ w1, const float* __restrict__ b1,
    const float* __restrict__ w2, const float* __restrict__ b2,
    const float* __restrict__ w3, const float* __restrict__ b3v,
    const float* __restrict__ tmaxb,
    float* __restrict__ out, int n)
{
    const int lane = threadIdx.x & 31;
    const int hi   = lane >> 4;       // 0: lanes 0-15, 1: lanes 16-31
    const int m    = lane & 15;       // event slot / N-column within fragment
    const long wave = (long)blockIdx.x * (blockDim.x >> 5) + (threadIdx.x >> 5);

    long ev    = wave * 16 + m;
    bool valid = (ev < n);
    long evc   = valid ? ev : (long)(n - 1);

    // ---- event load (lanes l and l+16 load the same event) ----
    float x = events[evc * 5 + 0];
    float y = events[evc * 5 + 1];
    float t = events[evc * 5 + 2];
    float p = events[evc * 5 + 3];
    int   bi = (int)events[evc * 5 + 4];

    float tn  = t / tmaxb[bi];
    int  p01  = (p > 0.f) ? 1 : 0;
    int  idxb = (int)x + WW * (int)y + WHC * p01 + WHC * 2 * bi;

    // ---- preload W2 as 16 B-fragments (4x16 f32 each, K = v + 2*hi) ----
    v2f bfrag[8][2];
    #pragma unroll
    for (int k0 = 0; k0 < 8; k0++) {
        #pragma unroll
        for (int nt = 0; nt < 2; nt++) {
            #pragma unroll
            for (int v = 0; v < 2; v++) {
                int K = 4 * k0 + v + 2 * hi;
                int N = nt * 16 + m;
                bfrag[k0][nt][v] = (K < HID && N < HID) ? w2[K * HID + N] : 0.f;
            }
        }
    }
    // ---- preload layer-1 weights matching the A-fragment K layout ----
    v2f w1v[8], b1v[8];
    #pragma unroll
    for (int k0 = 0; k0 < 8; k0++) {
        #pragma unroll
        for (int v = 0; v < 2; v++) {
            int j = 4 * k0 + v + 2 * hi;
            w1v[k0][v] = (j < HID) ? w1[j] : 0.f;
            b1v[k0][v] = (j < HID) ? b1[j] : 0.f;
        }
    }
    // ---- layer-3 constants (lane owns neuron m and m+16) ----
    float b2a = b2[m];
    float w3a = w3[m];
    float b2b = (m + 16 < HID) ? b2[m + 16] : 0.f;
    float w3b = (m + 16 < HID) ? w3[m + 16] : 0.f;
    float b3s = b3v[0];

    // ---- scatter operands: lane with (m<8) handles event (m + 8*hi) ----
    int   src    = (m & 7) + 8 * hi;
    float tn_sc  = __shfl(tn, src);
    int   idx_sc = __shfl(idxb, src);
    int   ok_sc  = __shfl((int)valid, src);
    float tnb3   = tn_sc * b3s;

    #pragma unroll
    for (int i = 0; i < CB; i++) {
        float z = tn - (float)i * 0.125f;
        v8f acc0 = {}, acc1 = {};
        #pragma unroll
        for (int k0 = 0; k0 < 8; k0++) {
            v2f afr;                                   // A fragment: h[m][K]
            afr[0] = fmaxf(z * w1v[k0][0] + b1v[k0][0], 0.f);
            afr[1] = fmaxf(z * w1v[k0][1] + b1v[k0][1], 0.f);
            acc0 = __builtin_amdgcn_wmma_f32_16x16x4_f32(
                       false, afr, false, bfrag[k0][0], (short)0, acc0, false, false);
            acc1 = __builtin_amdgcn_wmma_f32_16x16x4_f32(
                       false, afr, false, bfrag[k0][1], (short)0, acc1, false, false);
        }
        // layer 3: relu(acc + b2) dot w3, partial per lane (its two neurons)
        float s[8];
        #pragma unroll
        for (int r = 0; r < 8; r++)
            s[r] = fmaxf(acc0[r] + b2a, 0.f) * w3a +
                   fmaxf(acc1[r] + b2b, 0.f) * w3b;

        // tree-fold butterfly within each 16-lane half:
        // stage k reduces across lane-bit k and binds value-bit k to lane-bit k.
        // After 4 stages lane l holds sum_half(s[l&7]) -> exactly its event.
        #pragma unroll
        for (int r = 0; r < 8; r++) s[r] += __shfl_xor(s[r], 1);
        float u0 = (m & 1) ? s[1] : s[0];
        float u1 = (m & 1) ? s[3] : s[2];
        float u2 = (m & 1) ? s[5] : s[4];
        float u3 = (m & 1) ? s[7] : s[6];
        u0 += __shfl_xor(u0, 2);
        u1 += __shfl_xor(u1, 2);
        u2 += __shfl_xor(u2, 2);
        u3 += __shfl_xor(u3, 2);
        float v0 = (m & 2) ? u1 : u0;
        float v1 = (m & 2) ? u3 : u2;
        v0 += __shfl_xor(v0, 4);
        v1 += __shfl_xor(v1, 4);
        float f = (m & 4) ? v1 : v0;
        f += __shfl_xor(f, 8);

        if (m < 8 && ok_sc) {
            atomicAdd(out + idx_sc + WH * i, fmaf(tn_sc, f, tnb3));
        }
    }
}

// ---------------------------------------------------------------------------
extern "C" void kernel_launch(void* const* d_in, const int* in_sizes, int n_in,
                              void* d_out, int out_size, void* d_ws, size_t ws_size,
                              hipStream_t stream) {
    const float* events = (const float*)d_in[0];
    const float* w1 = (const float*)d_in[1];
    const float* b1 = (const float*)d_in[2];
    const float* w2 = (const float*)d_in[3];
    const float* b2 = (const float*)d_in[4];
    const float* w3 = (const float*)d_in[5];
    const float* b3 = (const float*)d_in[6];
    float*       out = (float*)d_out;
    int n = in_sizes[0] / 5;

    hipMemsetAsync(d_ws, 0, 4 * sizeof(unsigned), stream);
    hipMemsetAsync(d_out, 0, (size_t)out_size * sizeof(float), stream);

    est_tmax_kernel<<<512, 256, 0, stream>>>(events, n, (unsigned*)d_ws);

    long waves  = ((long)n + 15) / 16;
    int  blocks = (int)((waves + 7) / 8);       // 8 waves (256 threads) / block
    est_main_kernel<<<blocks, 256, 0, stream>>>(events, w1, b1, w2, b2, w3, b3,
                                                (const float*)d_ws, out, n);
}